// LatentQuantize_1726576854530
// MI455X (gfx1250) — compile-verified
//
#include <hip/hip_runtime.h>
#include <hip/hip_bf16.h>

#define DIMK   1024
#define NTOK   4096      // tokens per batch
#define BATCH  8
#define CB     6
#define OUT_ELEMS   (BATCH * DIMK * NTOK)        // 33554432
#define IDX_ELEMS   (BATCH * NTOK)               // 32768
#define N_TOKENS    (BATCH * NTOK)               // 32768
#define DEC_BLOCKS  16384                        // 8 waves/block, 131072 tiles

typedef __attribute__((ext_vector_type(2))) float v2f;
typedef __attribute__((ext_vector_type(8))) float v8f;

// ---------------------------------------------------------------------------
// Kernel 1: encode.  x = z^T @ Wi^T + bi ; quantize to grid j/8-0.5 ; emit
// q codes (padded 8 floats/token) to workspace and int32 indices to d_out.
// One thread per token; z loads coalesced along n; Wi kept in LDS transposed.
// ---------------------------------------------------------------------------
__global__ __launch_bounds__(256) void fsq_encode(
    const float* __restrict__ z, const float* __restrict__ Wi,
    const float* __restrict__ bi, float* __restrict__ q_ws,
    int* __restrict__ indices_out)
{
    __shared__ float WiT[DIMK * 8];   // padded row of 8 floats per d (32 KB)
    for (int i = threadIdx.x; i < CB * DIMK; i += 256) {
        int c = i >> 10;
        int d = i & (DIMK - 1);
        WiT[d * 8 + c] = Wi[i];
    }
    __syncthreads();

    const int token = blockIdx.x * 256 + threadIdx.x;   // grid = 128 blocks
    const int b = token >> 12;
    const int n = token & (NTOK - 1);
    const float* zp = z + (size_t)b * DIMK * NTOK + n;

    float a0 = 0.f, a1 = 0.f, a2 = 0.f, a3 = 0.f, a4 = 0.f, a5 = 0.f;
    for (int d = 0; d < DIMK; ++d) {
        float zv = zp[(size_t)d * NTOK];
        __builtin_prefetch(zp + (size_t)(d + 16) * NTOK, 0, 0); // strided stream
        const float4 w0 = *(const float4*)&WiT[d * 8];
        const float2 w1 = *(const float2*)&WiT[d * 8 + 4];
        a0 = fmaf(zv, w0.x, a0);
        a1 = fmaf(zv, w0.y, a1);
        a2 = fmaf(zv, w0.z, a2);
        a3 = fmaf(zv, w0.w, a3);
        a4 = fmaf(zv, w1.x, a4);
        a5 = fmaf(zv, w1.y, a5);
    }

    float acc[CB] = {a0, a1, a2, a3, a4, a5};
    float q[CB];
    int index = 0, p = 1;
#pragma unroll
    for (int c = 0; c < CB; ++c) {
        float x  = acc[c] + bi[c];
        float jf = rintf((x + 0.5f) * 8.0f);     // nearest grid level
        jf = fminf(7.0f, fmaxf(0.0f, jf));
        q[c] = jf * 0.125f - 0.5f;               // codes == quantized value
        index += (int)jf * p;                    // sum j_c * 8^c (exact)
        p *= 8;
    }
    float* qr = q_ws + (size_t)token * 8;
    *(float4*)qr       = make_float4(q[0], q[1], q[2], q[3]);
    *(float2*)(qr + 4) = make_float2(q[4], q[5]);
    indices_out[token] = index;
}

// ---------------------------------------------------------------------------
// Kernel 2: decode + fused loss.  out[d,t] = sum_c Wo[d,c] * q[t,c] + bo[d]
// One wave computes a 16(d) x 16(token) tile with V_WMMA_F32_16X16X4_F32,
// K=6 padded to two K=4 steps.  f32 A(16x4): lane&15=M, lane>>4 -> K+2,
// vgpr -> K lsb.  B(4x16): lane&15=N, lane>>4 -> K+2, vgpr -> K lsb.
// D: lane&15=N(token), vgpr r + 8*(lane>>4) = M(d)  ->  64B-contiguous stores.
// ---------------------------------------------------------------------------
__global__ __launch_bounds__(256) void fsq_decode(
    const float* __restrict__ z, const float* __restrict__ Wo,
    const float* __restrict__ bo, const float* __restrict__ q_ws,
    float* __restrict__ out, float* __restrict__ partial)
{
    const int wave = threadIdx.x >> 5;
    const int lane = threadIdx.x & 31;
    const int wid  = blockIdx.x * 8 + wave;       // 131072 waves total
    const int t_tile =  wid        & 255;         // 256 token tiles
    const int d_tile = (wid >> 8)  & 63;          // 64 d tiles
    const int b      =  wid >> 14;                // 8 batches

    const int hi  = lane >> 4;
    const int l15 = lane & 15;

    // A = Wo tile (M=d rows, K=c), padded K=6 -> 8
    const float* wrow = Wo + (size_t)(d_tile * 16 + l15) * CB;
    const float w4 = wrow[4], w5 = wrow[5];
    v2f A0, A1;
    A0.x = wrow[0 + 2 * hi];  A0.y = wrow[1 + 2 * hi];   // K = {0,1} / {2,3}
    A1.x = hi ? 0.f : w4;     A1.y = hi ? 0.f : w5;      // K = {4,5} / {6,7}=0

    // B = q^T tile (K=c, N=token)
    const int tok = b * NTOK + t_tile * 16 + l15;
    const float* qrow = q_ws + (size_t)tok * 8;
    const float q4 = qrow[4], q5 = qrow[5];
    v2f B0, B1;
    B0.x = qrow[0 + 2 * hi];  B0.y = qrow[1 + 2 * hi];
    B1.x = hi ? 0.f : q4;     B1.y = hi ? 0.f : q5;

    v8f C = {};
    C = __builtin_amdgcn_wmma_f32_16x16x4_f32(false, A0, false, B0,
                                              (short)0, C, false, false);
    C = __builtin_amdgcn_wmma_f32_16x16x4_f32(false, A1, false, B1,
                                              (short)0, C, false, false);

    // Epilogue: + bo, store out, accumulate (out - z)^2
    float lsum = 0.f;
    const int tt = t_tile * 16 + l15;
    const size_t base = (size_t)b * DIMK * NTOK;
#pragma unroll
    for (int r = 0; r < 8; ++r) {
        int dd = d_tile * 16 + r + 8 * hi;
        size_t addr = base + (size_t)dd * NTOK + tt;
        float ov = C[r] + bo[dd];
        float zv = z[addr];
        out[addr] = ov;
        float df = ov - zv;
        lsum = fmaf(df, df, lsum);
    }

    // wave reduce (wave32) then block reduce -> per-block partial (deterministic)
#pragma unroll
    for (int o = 16; o > 0; o >>= 1) lsum += __shfl_xor(lsum, o, 32);
    __shared__ float wsum[8];
    if (lane == 0) wsum[wave] = lsum;
    __syncthreads();
    if (threadIdx.x == 0) {
        float s = 0.f;
#pragma unroll
        for (int i = 0; i < 8; ++i) s += wsum[i];
        partial[blockIdx.x] = s;
    }
}

// ---------------------------------------------------------------------------
// Kernel 3: fixed-order final reduction -> loss = 0.2 * mean((out - z)^2)
// ---------------------------------------------------------------------------
__global__ __launch_bounds__(256) void fsq_loss(
    const float* __restrict__ partial, int nparts, float* __restrict__ loss)
{
    __shared__ float s[256];
    float acc = 0.f;
    for (int i = threadIdx.x; i < nparts; i += 256) acc += partial[i];
    s[threadIdx.x] = acc;
    __syncthreads();
    for (int o = 128; o > 0; o >>= 1) {
        if (threadIdx.x < o) s[threadIdx.x] += s[threadIdx.x + o];
        __syncthreads();
    }
    if (threadIdx.x == 0) *loss = s[0] * (0.2f / (float)OUT_ELEMS);
}

// ---------------------------------------------------------------------------
extern "C" void kernel_launch(void* const* d_in, const int* in_sizes, int n_in,
                              void* d_out, int out_size, void* d_ws, size_t ws_size,
                              hipStream_t stream) {
    const float* z  = (const float*)d_in[0];   // (8,1024,4096)
    const float* Wi = (const float*)d_in[1];   // (6,1024)
    const float* bi = (const float*)d_in[2];   // (6,)
    const float* Wo = (const float*)d_in[3];   // (1024,6)
    const float* bo = (const float*)d_in[4];   // (1024,)
    // d_in[5] (vals) is a uniform grid; folded into closed-form quantizer.

    float* out     = (float*)d_out;                         // (8,1024,4096) f32
    int*   indices = (int*)((float*)d_out + OUT_ELEMS);     // (8,4096) i32
    float* loss    = (float*)d_out + OUT_ELEMS + IDX_ELEMS; // scalar f32

    float* q_ws    = (float*)d_ws;                          // 32768*8 f32 = 1 MiB
    float* partial = q_ws + (size_t)N_TOKENS * 8;           // 16384 f32

    fsq_encode<<<N_TOKENS / 256, 256, 0, stream>>>(z, Wi, bi, q_ws, indices);
    fsq_decode<<<DEC_BLOCKS, 256, 0, stream>>>(z, Wo, bo, q_ws, out, partial);
    fsq_loss<<<1, 256, 0, stream>>>(partial, DEC_BLOCKS, loss);
}